// Upsampler_1254130451006
// MI455X (gfx1250) — compile-verified
//
#include <hip/hip_runtime.h>

// ROIAlign-style crop + bilinear upsample to 299x299, expressed per 16x16
// output tile as OUT = R * P * C  (R,C = tiny interpolation-weight matrices,
// P = dense input patch), executed with V_WMMA_F32_16X16X4_F32.
//
// Memory-bound problem (549 MB stores @ 23.3 TB/s ~ 24us floor); WMMA path
// keeps stores coalesced, makes input reads dense/L2-resident, and cuts VALU
// work ~3-4x vs scalar lerp.

#define OUT_N 299
#define TILE  16
#define NT    19          // ceil(299/16)
#define WAVES 4           // waves per block (128 threads, wave32)

typedef __attribute__((ext_vector_type(2))) float v2f;
typedef __attribute__((ext_vector_type(8))) float v8f;

// Interpolation taps for output index q along an axis:
//   src = clip((q+0.5)*len/299 - 0.5, 0, len-1) + lo ; i0=floor ; i1=min(i0+1,hi-1)
// Returned as patch-local indices (k = i - base) and weights.
__device__ __forceinline__ void edge_taps(int q, int lo, int hi, float len, int base,
                                          int& k0, int& k1, float& w0, float& w1)
{
    float s = (q + 0.5f) * len / 299.0f - 0.5f;
    s = fminf(fmaxf(s, 0.0f), len - 1.0f) + (float)lo;
    float f0 = floorf(s);
    float w  = s - f0;
    int i0 = (int)f0;
    int i1 = min(i0 + 1, hi - 1);
    k0 = i0 - base;
    k1 = i1 - base;
    w0 = 1.0f - w;
    w1 = w;
}

// patch-local base index for a tile starting at output index qbase
__device__ __forceinline__ int axis_base(int qbase, int lo, float len)
{
    float s = (qbase + 0.5f) * len / 299.0f - 0.5f;
    s = fminf(fmaxf(s, 0.0f), len - 1.0f) + (float)lo;
    int b = (int)floorf(s);
    return min(b, OUT_N - TILE);   // keep the 16-wide patch fully in-bounds
}

__global__ __launch_bounds__(WAVES * 32)
void roi_upsample_wmma(const float* __restrict__ x, const int* __restrict__ f,
                       float* __restrict__ out, int S, int G, long long total_tiles)
{
    // per-wave, per-channel LDS slot: 16 rows padded to 17 floats (bank-conflict free)
    __shared__ float smem[WAVES * 3 * (TILE * 17)];

    const int lane = threadIdx.x & 31;
    const int w    = threadIdx.x >> 5;
    const int half = lane >> 4;       // 0: lanes 0-15, 1: lanes 16-31
    const int m    = lane & 15;       // matrix row/col index held by this lane

    long long tile = (long long)blockIdx.x * WAVES + w;
    if (tile >= total_tiles) tile = total_tiles - 1;   // idempotent tail (keeps barriers convergent)

    const int t  = (int)(tile % (NT * NT));
    const int sg = (int)(tile / (NT * NT));
    const int ty = t / NT, tx = t % NT;
    const int s  = sg / G, g  = sg % G;
    const int ibase = ty * TILE;
    const int jbase = tx * TILE;

    const int* box = f + (size_t)(s * G + g) * 4;
    const int tlx = box[0], tly = box[1], brx = box[2], bry = box[3];
    const float hcf = (float)(brx - tlx);   // crop extent along output rows
    const float wcf = (float)(bry - tly);   // crop extent along output cols

    const int rbase = axis_base(ibase, tlx, hcf);
    const int cbase = axis_base(jbase, tly, wcf);

    // taps for this lane's output row/col (clamped for partial edge tiles; masked at store)
    const int i = min(ibase + m, OUT_N - 1);
    const int j = min(jbase + m, OUT_N - 1);
    int k0r, k1r; float w0r, w1r;
    int k0c, k1c; float w0c, w1c;
    edge_taps(i, tlx, brx, hcf, rbase, k0r, k1r, w0r, w1r);
    edge_taps(j, tly, bry, wcf, cbase, k0c, k1c, w0c, w1c);

    // A-operand fragments (16x4 f32 per call): lane holds row M=m,
    // a.x/a.y = K = 4*k4 + 2*half, +1.
    v2f aC[4];   // GEMM1: A = C^T  (row = output col, K = patch col)
    v2f aR[4];   // GEMM2: A = R    (row = output row, K = patch row)
#pragma unroll
    for (int k4 = 0; k4 < 4; ++k4) {
        const int kx = 4 * k4 + 2 * half;
        aC[k4].x = (kx     == k0c ? w0c : 0.0f) + (kx     == k1c ? w1c : 0.0f);
        aC[k4].y = (kx + 1 == k0c ? w0c : 0.0f) + (kx + 1 == k1c ? w1c : 0.0f);
        aR[k4].x = (kx     == k0r ? w0r : 0.0f) + (kx     == k1r ? w1r : 0.0f);
        aR[k4].y = (kx + 1 == k0r ? w0r : 0.0f) + (kx + 1 == k1r ? w1r : 0.0f);
    }

#pragma unroll
    for (int c = 0; c < 3; ++c) {
        // ---- GEMM1: S = C^T * P^T  (S[j][r] = (P*C)[r][j]) ----
        // B-operand: B[k][n] = P[n][k] -> lane n=m reads image row rbase+m,
        // two consecutive columns cbase+kx, +1.
        const float* img = x + ((size_t)(s * 3 + c) * OUT_N + (size_t)(rbase + m)) * OUT_N + cbase;
        v8f Sacc = (v8f)0.0f;
#pragma unroll
        for (int k4 = 0; k4 < 4; ++k4) {
            const int kx = 4 * k4 + 2 * half;
            v2f b; b.x = img[kx]; b.y = img[kx + 1];
            Sacc = __builtin_amdgcn_wmma_f32_16x16x4_f32(
                false, aC[k4], false, b, (short)0, Sacc, false, false);
        }

        // ---- transpose bounce through LDS (D-layout -> B-operand layout) ----
        float* sm = smem + (size_t)(w * 3 + c) * (TILE * 17);
#pragma unroll
        for (int v = 0; v < 8; ++v)
            sm[(v + 8 * half) * 17 + m] = Sacc[v];    // S row j = v+8*half, col r = m
        __syncthreads();

        // ---- GEMM2: OUT = R * (P*C);  B[k][n] = S[n][k] -> lane n=m reads sm row m ----
        v8f Oacc = (v8f)0.0f;
#pragma unroll
        for (int k4 = 0; k4 < 4; ++k4) {
            const int kx = 4 * k4 + 2 * half;
            v2f b; b.x = sm[m * 17 + kx]; b.y = sm[m * 17 + kx + 1];
            Oacc = __builtin_amdgcn_wmma_f32_16x16x4_f32(
                false, aR[k4], false, b, (short)0, Oacc, false, false);
        }

        // ---- coalesced masked stores (D-layout: row = v+8*half, col = m) ----
        float* op = out + (size_t)((s * G + g) * 3 + c) * OUT_N * OUT_N;
#pragma unroll
        for (int v = 0; v < 8; ++v) {
            const int oi = ibase + v + 8 * half;
            const int oj = jbase + m;
            if (oi < OUT_N && oj < OUT_N)
                op[(size_t)oi * OUT_N + oj] = Oacc[v];
        }
    }
}

extern "C" void kernel_launch(void* const* d_in, const int* in_sizes, int n_in,
                              void* d_out, int out_size, void* d_ws, size_t ws_size,
                              hipStream_t stream)
{
    const float* x = (const float*)d_in[0];   // (S, 3, 299, 299) f32
    const int*   f = (const int*)d_in[1];     // (S, G, 4) i32
    float* out = (float*)d_out;               // (S, G, 3, 299, 299) f32

    int S = in_sizes[0] / (3 * OUT_N * OUT_N);
    if (S < 1) S = 1;
    int G = in_sizes[1] / (S * 4);
    if (G < 1) G = 1;

    const long long total_tiles = (long long)S * G * NT * NT;    // 184832 for S=32,G=16
    const int blocks = (int)((total_tiles + WAVES - 1) / WAVES); // 46208

    roi_upsample_wmma<<<blocks, WAVES * 32, 0, stream>>>(x, f, out, S, G, total_tiles);
}